// Preprocessor_89378269430452
// MI455X (gfx1250) — compile-verified
//
#include <hip/hip_runtime.h>
#include <hip/hip_bf16.h>
#include <math.h>

// ---------------- problem constants ----------------
#define ZD   16
#define VDIM 64
#define NTR  32
#define TT   8192

#define LOG2PI_F 1.8378770664093453f

// workspace layout (in floats)
#define WS_Q      ((size_t)0)
#define WS_S0     ((size_t)256)
#define WS_LG     ((size_t)512)
#define WS_LGI    ((size_t)768)
#define WS_SCAL   ((size_t)1024)                 // c0, c1
#define WS_PROG   ((size_t)1032)                 // int progress counter
#define WS_LOGDET ((size_t)1056)                 // TT floats
#define WS_D      ((size_t)16384)                // TT*256 floats
#define WS_VT     (WS_D + (size_t)TT*256)        // NTR*VDIM*TT floats
#define WS_FULL_FLOATS  (WS_VT + (size_t)NTR*VDIM*TT)

typedef float v2f __attribute__((ext_vector_type(2)));
typedef float v8f __attribute__((ext_vector_type(8)));

// ---------------- WMMA helpers (wave32, fp32 16x16x4) ----------------
// A-matrix 16x4 f32 frag: lane l, vgpr j -> A[l%16][2*(l/16)+j]
// B-matrix 4x16 f32 frag: lane l, vgpr j -> B[2*(l/16)+j][l%16]
// C/D 16x16 f32: vgpr r, lane l -> (row r+8*(l/16), col l%16)
template<bool TA, bool TB>
static __device__ __forceinline__
v8f wmma_tile(const float* As, int lda, const float* Bs, int ldb, int K, v8f acc) {
    const int lane = threadIdx.x & 31;
    const int half = lane >> 4;
    const int m    = lane & 15;
#pragma unroll
    for (int kb = 0; kb < K; kb += 4) {
        const int ka = kb + 2 * half;
        v2f a, b;
        if (!TA) { a.x = As[m * lda + ka];       a.y = As[m * lda + ka + 1]; }
        else     { a.x = As[ka * lda + m];       a.y = As[(ka + 1) * lda + m]; }
        if (!TB) { b.x = Bs[ka * ldb + m];       b.y = Bs[(ka + 1) * ldb + m]; }
        else     { b.x = Bs[m * ldb + ka];       b.y = Bs[m * ldb + ka + 1]; }
        acc = __builtin_amdgcn_wmma_f32_16x16x4_f32(false, a, false, b,
                                                    (short)0, acc, false, false);
    }
    return acc;
}

static __device__ __forceinline__ void cd_store(float* Ds, int ldd, v8f d) {
    const int lane = threadIdx.x & 31, half = lane >> 4, n = lane & 15;
#pragma unroll
    for (int r = 0; r < 8; ++r) Ds[(r + 8 * half) * ldd + n] = d[r];
}

static __device__ __forceinline__ v8f cd_load(const float* Cs, int ldc) {
    const int lane = threadIdx.x & 31, half = lane >> 4, n = lane & 15;
    v8f c;
#pragma unroll
    for (int r = 0; r < 8; ++r) c[r] = Cs[(r + 8 * half) * ldc + n];
    return c;
}

// ---------------- kernel A: setup constants ----------------
__global__ __launch_bounds__(256) void setup_kernel(
        const float* __restrict__ Bm, const float* __restrict__ W,
        const float* __restrict__ S0h, const float* __restrict__ lsv,
        float* __restrict__ ws) {
    __shared__ float sG[256];
    const int tid = threadIdx.x;
    const int i = tid >> 4, j = tid & 15;

    float q = (i == j) ? 1e-6f : 0.0f;
#pragma unroll
    for (int k = 0; k < 16; ++k) if (k <= i && k <= j) q += Bm[i*16+k] * Bm[j*16+k];
    ws[WS_Q + tid] = q;

    float s0 = (i == j) ? 1e-6f : 0.0f;
#pragma unroll
    for (int k = 0; k < 16; ++k) if (k <= i && k <= j) s0 += S0h[i*16+k] * S0h[j*16+k];
    ws[WS_S0 + tid] = s0;

    float g = 0.0f;
#pragma unroll 8
    for (int k = 0; k < 64; ++k) g += W[k*16+i] * W[k*16+j];
    sG[tid] = g;

    if (tid == 0) {
        float sv = __expf(lsv[0]);
        float var_v = sv * sv;
        ws[WS_SCAL + 0] = var_v + 1e-6f;          // c0 (t==0)
        ws[WS_SCAL + 1] = var_v + 2e-6f;          // c1 (t>=1)
        *(int*)(ws + WS_PROG) = 0;                // reset producer progress
    }
    __syncthreads();

    // 16x16 Cholesky of G (lower), in LDS
    for (int k = 0; k < 16; ++k) {
        float dk = sqrtf(sG[k*16+k]);
        __syncthreads();
        if (tid == k*16+k) sG[tid] = dk;
        if (j == k && i > k) sG[i*16+k] = sG[i*16+k] / dk;
        __syncthreads();
        if (i > k && j > k && j <= i) sG[i*16+j] -= sG[i*16+k] * sG[j*16+k];
        __syncthreads();
    }
    float lg = (j > i) ? 0.0f : sG[tid];          // zero strict upper
    __syncthreads();
    sG[tid] = lg;
    ws[WS_LG + tid] = lg;
    __syncthreads();

    // LgInv: forward solve Lg x = e_r, column r per thread
    if (tid < 16) {
        const int r = tid;
        float x[16];
#pragma unroll
        for (int ii = 0; ii < 16; ++ii) {
            float s = (ii == r) ? 1.0f : 0.0f;
#pragma unroll
            for (int k2 = 0; k2 < 16; ++k2) if (k2 < ii) s -= sG[ii*16+k2] * x[k2];
            x[ii] = s / sG[ii*16+ii];
        }
#pragma unroll
        for (int ii = 0; ii < 16; ++ii) ws[WS_LGI + ii*16 + r] = x[ii];
    }
}

// ---------------- kernel T: transpose v[n,i,t] -> vt[t, n*64+i] ----------------
__global__ __launch_bounds__(256) void transpose_kernel(
        const float* __restrict__ in, float* __restrict__ out) {
    __shared__ float tile[32][33];
    const int t0 = blockIdx.x * 32;
    const int r0 = blockIdx.y * 32;
    for (int k = threadIdx.y; k < 32; k += 8)
        tile[k][threadIdx.x] = in[(size_t)(r0 + k) * TT + t0 + threadIdx.x];
    __syncthreads();
    for (int k = threadIdx.y; k < 32; k += 8)
        out[(size_t)(t0 + k) * (NTR * VDIM) + r0 + threadIdx.x] = tile[threadIdx.x][k];
}

// ---------------- producer step (covariance recursion) -----------------------
template<bool FIRST>
static __device__ __forceinline__ void cov_iter(
        int t, float ct, int lane, const float* gS0,
        float* sA, float* sLg, float* sLgI, float* sQ,
        float* sSf, float* sSd, float* sT1, float* sP,
        float* sLpI, float* sU1, float* sU2, float* sTmp,
        float* gD, float* gLd, int* prog) {
    if (FIRST) {
        for (int i = lane; i < 256; i += 32) sSd[i] = gS0[i];
        __syncthreads();
    } else {
        // Tmp = A @ Sf ; Sd = sym(Tmp @ A^T + Q)
        v8f acc = {};
        acc = wmma_tile<false, false>(sA, 16, sSf, 16, 16, acc);
        cd_store(sTmp, 16, acc);
        __syncthreads();
        v8f accq = cd_load(sQ, 16);
        accq = wmma_tile<false, true>(sTmp, 16, sA, 16, 16, accq);
        cd_store(sSd, 16, accq);
        __syncthreads();
        float t8[8];
#pragma unroll
        for (int r = 0; r < 8; ++r) {
            int idx = r * 32 + lane, i = idx >> 4, j = idx & 15;
            t8[r] = 0.5f * (sSd[i*16+j] + sSd[j*16+i]);
        }
        __syncthreads();
#pragma unroll
        for (int r = 0; r < 8; ++r) sSd[r*32 + lane] = t8[r];
        __syncthreads();
    }

    // T1 = Sd @ Lg
    {
        v8f a1 = {};
        a1 = wmma_tile<false, false>(sSd, 16, sLg, 16, 16, a1);
        cd_store(sT1, 16, a1);
    }
    __syncthreads();

    // P = Lg^T @ T1 + ct*I
    {
        const int half = lane >> 4, n = lane & 15;
        v8f ci = {};
#pragma unroll
        for (int r = 0; r < 8; ++r) ci[r] = ((r + 8*half) == n) ? ct : 0.0f;
        ci = wmma_tile<true, false>(sLg, 16, sT1, 16, 16, ci);
        cd_store(sP, 16, ci);
    }
    __syncthreads();

    // 16x16 SPD Cholesky of P (lower), accumulate logdet
    float ld = 48.0f * __logf(ct);
    for (int k = 0; k < 16; ++k) {
        float dk = sqrtf(sP[k*16+k]);
        ld += 2.0f * __logf(dk);
        if (lane == k) sP[k*16+k] = dk;
        if (lane > k && lane < 16) sP[lane*16+k] = sP[lane*16+k] / dk;
        __syncthreads();
        if (lane > k && lane < 16) {
            float lik = sP[lane*16+k];
            for (int j = k + 1; j <= lane; ++j) sP[lane*16+j] -= lik * sP[j*16+k];
        }
        __syncthreads();
    }
    if (lane == 0) gLd[t] = ld;

    // LpInv = Lp^-1 (triangular inversion, column r per lane)
    if (lane < 16) {
        const int r = lane;
        float x[16];
#pragma unroll
        for (int i = 0; i < 16; ++i) {
            if (i < r)       x[i] = 0.0f;
            else if (i == r) x[i] = 1.0f / sP[i*16+i];
            else {
                float s = 0.0f;
#pragma unroll
                for (int k2 = 0; k2 < 16; ++k2)
                    if (k2 >= r && k2 < i) s -= sP[i*16+k2] * x[k2];
                x[i] = s / sP[i*16+i];
            }
        }
#pragma unroll
        for (int i = 0; i < 16; ++i) sLpI[i*16+r] = x[i];
    }
    __syncthreads();

    // D = (T1 @ LpInv^T) @ (LpInv @ LgInv)
    {
        v8f u2 = {};
        u2 = wmma_tile<false, false>(sLpI, 16, sLgI, 16, 16, u2);
        cd_store(sU2, 16, u2);
        v8f u1 = {};
        u1 = wmma_tile<false, true>(sT1, 16, sLpI, 16, 16, u1);
        cd_store(sU1, 16, u1);
    }
    __syncthreads();
    {
        v8f ad = {};
        ad = wmma_tile<false, false>(sU1, 16, sU2, 16, 16, ad);
        cd_store(sTmp, 16, ad);
    }
    __syncthreads();

    // symmetrize, publish D_t, keep Sf = ct * D_t
#pragma unroll
    for (int r = 0; r < 8; ++r) {
        int idx = r * 32 + lane, i = idx >> 4, j = idx & 15;
        float dsym = 0.5f * (sTmp[i*16+j] + sTmp[j*16+i]);
        gD[(size_t)t * 256 + idx] = dsym;
        sSf[idx] = ct * dsym;
    }
    __syncthreads();
    if (lane == 0)
        __hip_atomic_store(prog, t + 1, __ATOMIC_RELEASE, __HIP_MEMORY_SCOPE_AGENT);
}

// ---------------- consumer step (mean recursion + LL) ------------------------
#define LDXC 65
template<bool FIRST>
static __device__ __forceinline__ void mean_iter(
        int t, float ct, int lane, int nb0, int use_vt,
        const float* v, const float* vt, const float* mu0,
        float* cA, float* cW,
        float* cMuf, float* cMud, float* cU, float* cE,
        float* cDm, float* cDt,
        const float* gD, const float* gLd, float& ll) {
    // prefetch x_t into registers (independent of producer)
    float4 xr[8];
    float  xs[32];
    if (use_vt) {
        const float* xt = vt + (size_t)t * (NTR * VDIM) + (size_t)nb0 * VDIM;
#pragma unroll
        for (int c4 = 0; c4 < 8; ++c4)
            xr[c4] = *(const float4*)(xt + (c4 * 32 + lane) * 4);
    } else {
#pragma unroll 4
        for (int s = 0; s < 32; ++s) {
            int f = s * 32 + lane;
            xs[s] = v[((size_t)(nb0 + (f >> 6)) * VDIM + (f & 63)) * TT + t];
        }
    }

    // load D_t
    for (int i = lane; i < 256; i += 32) cDt[i] = gD[(size_t)t * 256 + i];

    // 1. mud (diffused mean); first step -> broadcast mu0
    if (FIRST) {
        if (lane < 16) {
            float m0 = mu0[lane];
#pragma unroll
            for (int j = 0; j < 16; ++j) cMud[j * 17 + lane] = m0;
        }
    } else {
        v8f acc = {};
        acc = wmma_tile<false, true>(cMuf, 17, cA, 16, 16, acc);
        cd_store(cMud, 17, acc);
    }
    __syncthreads();

    // 2. m = mud @ W^T  (16 x 64)
#pragma unroll
    for (int nb = 0; nb < VDIM; nb += 16) {
        v8f acc = {};
        acc = wmma_tile<false, true>(cMud, 17, cW + nb * 16, 16, 16, acc);
        cd_store(cDm + nb, LDXC, acc);
    }
    __syncthreads();

    // 3. d = x_t - m (in place in cDm)
    if (use_vt) {
#pragma unroll
        for (int c4 = 0; c4 < 8; ++c4) {
            int f = (c4 * 32 + lane) * 4;
            int j = f >> 6, i0 = f & 63;
            float* dr = cDm + j * LDXC + i0;
            dr[0] = xr[c4].x - dr[0]; dr[1] = xr[c4].y - dr[1];
            dr[2] = xr[c4].z - dr[2]; dr[3] = xr[c4].w - dr[3];
        }
    } else {
#pragma unroll 4
        for (int s = 0; s < 32; ++s) {
            int f = s * 32 + lane;
            float* dp = cDm + (f >> 6) * LDXC + (f & 63);
            *dp = xs[s] - *dp;
        }
    }
    __syncthreads();

    // 4. u = d @ W  (K = 64, split into 4 independent chains)
    {
        v8f a0 = {}, a1 = {}, a2 = {}, a3 = {};
        a0 = wmma_tile<false, false>(cDm +  0, LDXC, cW +  0 * 16, 16, 16, a0);
        a1 = wmma_tile<false, false>(cDm + 16, LDXC, cW + 16 * 16, 16, 16, a1);
        a2 = wmma_tile<false, false>(cDm + 32, LDXC, cW + 32 * 16, 16, 16, a2);
        a3 = wmma_tile<false, false>(cDm + 48, LDXC, cW + 48 * 16, 16, 16, a3);
        v8f u = (a0 + a1) + (a2 + a3);
        cd_store(cU, 17, u);
    }
    __syncthreads();

    // 5. e = u @ D_t
    {
        v8f acc = {};
        acc = wmma_tile<false, false>(cU, 17, cDt, 16, 16, acc);
        cd_store(cE, 17, acc);
    }
    __syncthreads();

    // 6. muf = mud + e
    for (int i = lane; i < 256; i += 32) {
        int j = i >> 4, k = i & 15;
        cMuf[j*17+k] = cMud[j*17+k] + cE[j*17+k];
    }

    // 7. LL contribution, trial = lane (lanes 0..15)
    if (lane < 16) {
        float sd2 = 0.0f;
#pragma unroll 8
        for (int i = 0; i < VDIM; ++i) { float dv = cDm[lane*LDXC + i]; sd2 += dv * dv; }
        float ue = 0.0f;
#pragma unroll
        for (int i = 0; i < 16; ++i) ue += cU[lane*17+i] * cE[lane*17+i];
        float quad = (sd2 - ue) / ct;
        ll += -0.5f * VDIM * LOG2PI_F - 0.5f * gLd[t] - 0.5f * quad;
    }
    __syncthreads();
}

// ---------------- fused producer/consumer kernel -----------------------------
// grid = 3 blocks x 1 wave.
// block 0 (producer): covariance recursion, publishes D_t / logdet_t + progress.
// blocks 1..2 (consumers): mean recursion + LL for 16 trials each; they poll
// the progress counter with agent-scope acquire loads only when behind.
__global__ __launch_bounds__(32) void fused_kernel(
        const float* __restrict__ v, const float* __restrict__ Amat,
        const float* __restrict__ W, const float* __restrict__ mu0,
        float* ws, float* __restrict__ out, int use_vt) {
    const int lane = threadIdx.x;
    float* gD  = ws + WS_D;
    float* gLd = ws + WS_LOGDET;
    int*   prog = (int*)(ws + WS_PROG);
    const float c0 = ws[WS_SCAL + 0];
    const float c1 = ws[WS_SCAL + 1];

    if (blockIdx.x == 0) {
        // =================== PRODUCER ===================
        __shared__ float sA[256], sLg[256], sLgI[256], sQ[256];
        __shared__ float sSf[256], sSd[256], sT1[256], sP[256];
        __shared__ float sLpI[256], sU1[256], sU2[256], sTmp[256];

        for (int i = lane; i < 256; i += 32) {
            sA[i]   = Amat[i];
            sQ[i]   = ws[WS_Q + i];
            sLg[i]  = ws[WS_LG + i];
            sLgI[i] = ws[WS_LGI + i];
        }
        __syncthreads();

        cov_iter<true>(0, c0, lane, ws + WS_S0, sA, sLg, sLgI, sQ,
                       sSf, sSd, sT1, sP, sLpI, sU1, sU2, sTmp, gD, gLd, prog);
        for (int t = 1; t < TT; ++t)
            cov_iter<false>(t, c1, lane, ws + WS_S0, sA, sLg, sLgI, sQ,
                            sSf, sSd, sT1, sP, sLpI, sU1, sU2, sTmp, gD, gLd, prog);
    } else {
        // =================== CONSUMERS ===================
        __shared__ float cA[256], cW[VDIM * 16];
        __shared__ float cMuf[16 * 17], cMud[16 * 17], cU[16 * 17], cE[16 * 17];
        __shared__ float cDm[16 * LDXC], cDt[256];
        const int nb0 = (blockIdx.x - 1) * 16;    // first trial of this block

        for (int i = lane; i < 256; i += 32) cA[i] = Amat[i];
        for (int i = lane; i < VDIM * 16; i += 32) cW[i] = W[i];
        const float* vt = ws + WS_VT;
        float ll = 0.0f;
        int avail = 0;
        __syncthreads();

        // wait for step 0, then peeled first iteration
        while (avail < 1) {
            avail = __hip_atomic_load(prog, __ATOMIC_ACQUIRE, __HIP_MEMORY_SCOPE_AGENT);
            if (avail < 1) __builtin_amdgcn_s_sleep(2);
        }
        mean_iter<true>(0, c0, lane, nb0, use_vt, v, vt, mu0,
                        cA, cW, cMuf, cMud, cU, cE, cDm, cDt, gD, gLd, ll);

        for (int t = 1; t < TT; ++t) {
            // only touch the atomic (and pay the L0 invalidate) when behind
            while (avail < t + 1) {
                avail = __hip_atomic_load(prog, __ATOMIC_ACQUIRE, __HIP_MEMORY_SCOPE_AGENT);
                if (avail < t + 1) __builtin_amdgcn_s_sleep(2);
            }
            mean_iter<false>(t, c1, lane, nb0, use_vt, v, vt, mu0,
                             cA, cW, cMuf, cMud, cU, cE, cDm, cDt, gD, gLd, ll);
        }
        if (lane < 16) out[nb0 + lane] = ll;
    }
}

// ---------------- launch ----------------
extern "C" void kernel_launch(void* const* d_in, const int* in_sizes, int n_in,
                              void* d_out, int out_size, void* d_ws, size_t ws_size,
                              hipStream_t stream) {
    (void)in_sizes; (void)n_in; (void)out_size;
    const float* v    = (const float*)d_in[0];
    const float* Amat = (const float*)d_in[1];
    const float* Bm   = (const float*)d_in[2];
    const float* W    = (const float*)d_in[3];
    const float* mu0  = (const float*)d_in[4];
    const float* S0h  = (const float*)d_in[5];
    const float* lsv  = (const float*)d_in[6];
    float* ws  = (float*)d_ws;
    float* out = (float*)d_out;

    const int use_vt = (ws_size >= WS_FULL_FLOATS * sizeof(float)) ? 1 : 0;

    setup_kernel<<<1, 256, 0, stream>>>(Bm, W, S0h, lsv, ws);
    if (use_vt) {
        transpose_kernel<<<dim3(TT / 32, (NTR * VDIM) / 32), dim3(32, 8), 0, stream>>>(
            v, ws + WS_VT);
    }
    fused_kernel<<<3, 32, 0, stream>>>(v, Amat, W, mu0, ws, out, use_vt);
}